// UpDownHierarchyAttentionCore_67731634258423
// MI455X (gfx1250) — compile-verified
//
#include <hip/hip_runtime.h>
#include <math.h>

typedef __attribute__((ext_vector_type(2))) float v2f;
typedef __attribute__((ext_vector_type(8))) float v8f;
typedef __attribute__((ext_vector_type(4))) unsigned int u32x4;
typedef __attribute__((ext_vector_type(8))) int i32x8;
typedef __attribute__((ext_vector_type(4))) int i32x4;

// Problem sizes (fixed by the reference)
#define BB 64
#define NN 36
#define MM 36
#define RR 1024
#define AA 512
#define EE 1024

// GEMM tiling
#define KC   32           // K-chunk staged per TDM transfer
#define RS   33           // LDS row stride in floats (32 + 1 pad via TDM pad_enable)
#define TILEF (16 * RS)   // floats per staged tile (528)

#if __has_builtin(__builtin_amdgcn_tensor_load_to_lds) && __has_builtin(__builtin_amdgcn_s_wait_tensorcnt)
#define USE_TDM 1
#else
#define USE_TDM 0
#endif

__device__ __forceinline__ float sigmoidf_(float x) { return 1.0f / (1.0f + __expf(-x)); }

#if USE_TDM
// Issue a TDM DMA of a 16-row x KC-col f32 tile of W (row stride ldW elems)
// into LDS at byte offset lds_off. Padding: +1 DWORD after every 32 DWORDs
// stored -> LDS row stride 33 floats (bank-conflict-free 16-row reads).
// D# layout per CDNA5 ISA (group0 + group1; groups 2/3 zero).
// This toolchain (clang-23 / therock-10.0) uses the 6-arg builtin form.
__device__ __forceinline__ void tdm_load_tile(const float* gptr, int ldW_elems,
                                              unsigned lds_off) {
  unsigned long long ga = (unsigned long long)(size_t)gptr;
  u32x4 g0;
  g0.x = 1u;                                               // count=1, user mode
  g0.y = lds_off;                                          // lds_addr (bytes)
  g0.z = (unsigned)(ga & 0xFFFFFFFFu);                     // global_addr[31:0]
  g0.w = (unsigned)((ga >> 32) & 0x01FFFFFFu) | (2u << 30);// addr[56:32] | type=2
  i32x8 g1;
  g1[0] = (int)((2u << 16) | (1u << 20) | (4u << 22));     // data_size=4B, pad_en, interval=32dw, amount=1dw
  g1[1] = 0;                                               // barrier addr=0; tensor_dim0[15:0]=0
  g1[2] = (int)0x00004000;                                 // tensor_dim0 = 1<<30 (no OOB clamp)
  g1[3] = (int)(0x4000u | ((unsigned)KC << 16));           // tensor_dim1 = 1<<30 ; tile_dim0 = KC
  g1[4] = 16;                                              // tile_dim1 = 16 rows; tile_dim2 = 0
  g1[5] = ldW_elems;                                       // tensor_dim0_stride[31:0]
  g1[6] = 0;                                               // stride hi; dim1_stride lo
  g1[7] = 0;
  i32x4 z4 = { 0, 0, 0, 0 };
  i32x8 z8 = { 0, 0, 0, 0, 0, 0, 0, 0 };
  __builtin_amdgcn_tensor_load_to_lds(g0, g1, z4, z4, z8, 0);
}
#endif

// ---------------------------------------------------------------------------
// Segmented GEMM: C[64, N] = sum_s A_s (64,K_s) @ W_s^T (N,K_s) + b1 + b2.
// One wave owns a 16-column tile and all four 16-row M tiles (weights read
// exactly once grid-wide). W tiles are TDM-DMA'd into double-buffered LDS
// (per-wave private region, TENSORcnt prefetch pipeline); A rows are small,
// L2-resident direct loads. Compute: V_WMMA_F32_16X16X4_F32.
// ---------------------------------------------------------------------------
struct Seg { const float* A; const float* W; int ldA; int ldW; int K; };
struct GArgs {
  Seg s[4];
  int nseg;
  const float* b1;
  const float* b2;
  float* C;          // (64, N) row-major
  int N;
  int act;           // 0 = none, 1 = sigmoid
};

__global__ __launch_bounds__(256) void gemm64_wmma_f32(GArgs ga) {
#if USE_TDM
  __shared__ float smem[8 * 2 * TILEF];
#endif
  const int lane = threadIdx.x & 31;
  const int wave = threadIdx.x >> 5;
  const int n0 = (blockIdx.x * 8 + wave) * 16;
  if (n0 >= ga.N) return;
  const int nl = lane & 15;   // n index (B frag) / m index (A frag)
  const int kh = lane >> 4;   // K-half selector per ISA 16x4 f32 layout

  v8f zero = {};
  v8f acc[4] = { zero, zero, zero, zero };

#if USE_TDM
  float* myLds = &smem[wave * 2 * TILEF];
  const unsigned ldsBase = (unsigned)(size_t)(void*)myLds;   // LDS byte offset (low 32 bits)
#endif

  for (int s = 0; s < ga.nseg; ++s) {
    const float* A = ga.s[s].A;
    const float* W = ga.s[s].W;
    const int ldA = ga.s[s].ldA;
    const int ldW = ga.s[s].ldW;
    const int K   = ga.s[s].K;

#if USE_TDM
    const float* wtile = W + (size_t)n0 * (size_t)ldW;   // rows n0..n0+15 of W
    tdm_load_tile(wtile, ldW, ldsBase);                  // prefetch chunk 0 -> buf 0
    int cur = 0;
    for (int k0 = 0; k0 < K; k0 += KC) {
      if (k0 + KC < K) {
        tdm_load_tile(wtile + (k0 + KC), ldW, ldsBase + (unsigned)((cur ^ 1) * TILEF * 4));
        __builtin_amdgcn_s_wait_tensorcnt(1);            // current chunk landed
      } else {
        __builtin_amdgcn_s_wait_tensorcnt(0);
      }
      const float* sb = myLds + cur * TILEF;
      for (int kk = 0; kk < KC; kk += 4) {
        const int ka = kk + 2 * kh;
        v2f bf = { sb[nl * RS + ka], sb[nl * RS + ka + 1] };
#pragma unroll
        for (int mt = 0; mt < 4; ++mt) {
          const float* ap = A + (size_t)(mt * 16 + nl) * (size_t)ldA + (k0 + ka);
          v2f af = { ap[0], ap[1] };
          acc[mt] = __builtin_amdgcn_wmma_f32_16x16x4_f32(
              false, af, false, bf, (short)0, acc[mt], false, false);
        }
      }
      cur ^= 1;
    }
#else
    const float* wrow = W + (size_t)(n0 + nl) * (size_t)ldW;
    for (int k0 = 0; k0 < K; k0 += 4) {
      const int ka = k0 + 2 * kh;
      v2f bf = { wrow[ka], wrow[ka + 1] };
#pragma unroll
      for (int mt = 0; mt < 4; ++mt) {
        const float* ap = A + (size_t)(mt * 16 + nl) * (size_t)ldA + ka;
        v2f af = { ap[0], ap[1] };
        acc[mt] = __builtin_amdgcn_wmma_f32_16x16x4_f32(
            false, af, false, bf, (short)0, acc[mt], false, false);
      }
    }
#endif
  }

  const int col = n0 + nl;
  float bias = 0.0f;
  if (ga.b1) bias += ga.b1[col];
  if (ga.b2) bias += ga.b2[col];
#pragma unroll
  for (int mt = 0; mt < 4; ++mt) {
#pragma unroll
    for (int r = 0; r < 8; ++r) {
      const int row = mt * 16 + r + 8 * kh;   // C/D layout: lanes16-31 -> M=r+8
      float v = acc[mt][r] + bias;
      if (ga.act == 1) v = sigmoidf_(v);
      ga.C[(size_t)row * (size_t)ga.N + col] = v;
    }
  }
}

// ---------------------------------------------------------------------------
// LSTM pointwise: gate order i,f,g,o. One thread per (b,d) of 64x1024.
// ---------------------------------------------------------------------------
__global__ __launch_bounds__(256) void lstm_pointwise(
    const float* __restrict__ G, const float* __restrict__ cprev,
    float* hA, float* hB, float* cA) {
  const int j = blockIdx.x * 256 + threadIdx.x;          // 0..65535
  const int b = j >> 10, d = j & 1023;
  const float* g = G + (size_t)b * 4096;
  const float i  = sigmoidf_(g[d]);
  const float f  = sigmoidf_(g[1024 + d]);
  const float gg = tanhf(g[2048 + d]);
  const float o  = sigmoidf_(g[3072 + d]);
  const float c  = f * cprev[j] + i * gg;
  const float h  = o * tanhf(c);
  hA[j] = h;
  if (hB) hB[j] = h;
  cA[j] = c;
}

// ---------------------------------------------------------------------------
// Attention core: scores = tanh(p + att_h) . aW + ab ; masked renorm softmax;
// weighted feature sum. One block per batch element. gidx!=null => gathered
// (B,N,M,*) tensors indexed by [b, gidx[b]].
// ---------------------------------------------------------------------------
__global__ __launch_bounds__(256) void attention_kernel(
    const float* __restrict__ atth,    // (64, 512)
    const float* __restrict__ feats,
    const float* __restrict__ p,
    const float* __restrict__ mask,
    const int*   __restrict__ gidx,    // null for att1
    const float* __restrict__ aW,      // (512)
    const float* __restrict__ aB,      // (1)
    float* __restrict__ out,           // (64, 1024)
    int* idxOut) {                     // (64) or null
  __shared__ float sh[AA];
  __shared__ float sw[64];
  const int b = blockIdx.x;
  const int t = threadIdx.x;
  const int lane = t & 31;
  const int wave = t >> 5;

  size_t fo, po, mo;
  if (gidx) {
    const size_t br = (size_t)b * NN + (size_t)gidx[b];
    fo = br * MM * RR; po = br * MM * AA; mo = br * MM;
  } else {
    fo = (size_t)b * NN * RR; po = (size_t)b * NN * AA; mo = (size_t)b * NN;
  }
  const float* F = feats + fo;
  const float* P = p + po;
  const float* Mk = mask + mo;

  for (int i = t; i < AA; i += 256) sh[i] = atth[(size_t)b * AA + i];
  __syncthreads();

  for (int n = wave; n < NN; n += 8) {
    const float* pr = P + (size_t)n * AA;
    float part = 0.0f;
    for (int a = lane; a < AA; a += 32) part += tanhf(pr[a] + sh[a]) * aW[a];
#pragma unroll
    for (int off = 16; off > 0; off >>= 1) part += __shfl_down(part, off, 32);
    if (lane == 0) sw[n] = part + aB[0];
  }
  __syncthreads();

  if (t == 0) {
    float mx = -1e30f;
    for (int n = 0; n < NN; ++n) mx = fmaxf(mx, sw[n]);
    float sum = 0.0f;
    for (int n = 0; n < NN; ++n) { float e = __expf(sw[n] - mx); sw[n] = e; sum += e; }
    float msum = 0.0f;
    for (int n = 0; n < NN; ++n) { sw[n] = (sw[n] / sum) * Mk[n]; msum += sw[n]; }
    const float inv = 1.0f / (msum + 1e-8f);
    int am = 0; float best = -1e30f;
    for (int n = 0; n < NN; ++n) {
      sw[n] *= inv;
      if (sw[n] > best) { best = sw[n]; am = n; }
    }
    if (idxOut) idxOut[b] = am;
  }
  __syncthreads();

  for (int d = t; d < RR; d += 256) {
    float s = 0.0f;
    for (int n = 0; n < NN; ++n) s += sw[n] * F[(size_t)n * RR + d];
    out[(size_t)b * RR + d] = s;
  }
}

// ---------------------------------------------------------------------------
// LN of g1*att2 and (1-g1)*att3 into contiguous concat buffer (64, 2048).
// ---------------------------------------------------------------------------
__global__ __launch_bounds__(256) void ln_gate_kernel(
    const float* __restrict__ g1, const float* __restrict__ att2,
    const float* __restrict__ att3,
    const float* __restrict__ ln1g, const float* __restrict__ ln1b,
    const float* __restrict__ ln2g, const float* __restrict__ ln2b,
    float* __restrict__ lncat) {
  __shared__ float red[256];
  const int b = blockIdx.x, t = threadIdx.x;
  for (int which = 0; which < 2; ++which) {
    float loc[4];
    float s = 0.0f;
    for (int i = 0; i < 4; ++i) {
      const int d = t + i * 256;
      const size_t j = (size_t)b * RR + d;
      const float g = g1[j];
      const float x = which ? (1.0f - g) * att3[j] : g * att2[j];
      loc[i] = x; s += x;
    }
    red[t] = s; __syncthreads();
    for (int off = 128; off > 0; off >>= 1) { if (t < off) red[t] += red[t + off]; __syncthreads(); }
    const float mu = red[0] / (float)RR;
    __syncthreads();
    float s2 = 0.0f;
    for (int i = 0; i < 4; ++i) { const float dd = loc[i] - mu; s2 += dd * dd; }
    red[t] = s2; __syncthreads();
    for (int off = 128; off > 0; off >>= 1) { if (t < off) red[t] += red[t + off]; __syncthreads(); }
    const float inv = rsqrtf(red[0] / (float)RR + 1e-6f);
    __syncthreads();
    const float* gg = which ? ln2g : ln1g;
    const float* gb = which ? ln2b : ln1b;
    for (int i = 0; i < 4; ++i) {
      const int d = t + i * 256;
      lncat[(size_t)b * 2048 + which * RR + d] = (loc[i] - mu) * inv * gg[d] + gb[d];
    }
    __syncthreads();
  }
}

__global__ __launch_bounds__(256) void combine_kernel(
    const float* __restrict__ g2, const float* __restrict__ att1,
    const float* __restrict__ att23, float* __restrict__ att) {
  const int j = blockIdx.x * 256 + threadIdx.x;  // 65536
  const float g = g2[j];
  att[j] = g * att1[j] + (1.0f - g) * att23[j];
}

// ---------------------------------------------------------------------------
extern "C" void kernel_launch(void* const* d_in, const int* in_sizes, int n_in,
                              void* d_out, int out_size, void* d_ws, size_t ws_size,
                              hipStream_t stream) {
  (void)in_sizes; (void)n_in; (void)out_size; (void)ws_size;
  const float* xt        = (const float*)d_in[0];
  const float* fc_feats  = (const float*)d_in[1];
  const float* att_feats = (const float*)d_in[2];
  const float* p_att     = (const float*)d_in[3];
  const float* h_state   = (const float*)d_in[4];   // (2,64,1024)
  const float* c_state   = (const float*)d_in[5];
  const float* att_masks = (const float*)d_in[6];
  const float* pos_feats = (const float*)d_in[7];
  const float* p_pos     = (const float*)d_in[8];
  const float* pos_mask  = (const float*)d_in[9];
  const float* sem_feats = (const float*)d_in[10];
  const float* p_sem     = (const float*)d_in[11];
  const float* sem_mask  = (const float*)d_in[12];
  const float* aWih = (const float*)d_in[13];  // (4096, 3072)
  const float* aWhh = (const float*)d_in[14];  // (4096, 1024)
  const float* abih = (const float*)d_in[15];
  const float* abhh = (const float*)d_in[16];
  const float* lWih = (const float*)d_in[17];  // (4096, 2048)
  const float* lWhh = (const float*)d_in[18];
  const float* lbih = (const float*)d_in[19];
  const float* lbhh = (const float*)d_in[20];
  const float* a1W = (const float*)d_in[21]; const float* a1b = (const float*)d_in[22];
  const float* a1aW = (const float*)d_in[23]; const float* a1ab = (const float*)d_in[24];
  const float* a2W = (const float*)d_in[25]; const float* a2b = (const float*)d_in[26];
  const float* a2aW = (const float*)d_in[27]; const float* a2ab = (const float*)d_in[28];
  const float* a3W = (const float*)d_in[29]; const float* a3b = (const float*)d_in[30];
  const float* a3aW = (const float*)d_in[31]; const float* a3ab = (const float*)d_in[32];
  const float* ln1g = (const float*)d_in[33]; const float* ln1b = (const float*)d_in[34];
  const float* ln2g = (const float*)d_in[35]; const float* ln2b = (const float*)d_in[36];
  const float* att23W = (const float*)d_in[37];
  const float* g1W = (const float*)d_in[38]; const float* g1b = (const float*)d_in[39];
  const float* g2W = (const float*)d_in[40]; const float* g2b = (const float*)d_in[41];

  float* out = (float*)d_out;
  // d_out: output(64k) | h_att(64k) h_lang(64k) | c_att(64k) c_lang(64k)
  float* o_hlang  = out;
  float* o_hatt_s = out + 65536;
  float* o_hlang_s= out + 131072;
  float* o_catt_s = out + 196608;
  float* o_clang_s= out + 262144;

  const float* h0 = h_state;
  const float* h1 = h_state + 65536;   // prev_h == h_state[-1]
  const float* c0 = c_state;
  const float* c1 = c_state + 65536;

  // Workspace layout (floats); ~4.2 MB total
  float* ws = (float*)d_ws;
  size_t off = 0;
  float* wG    = ws + off; off += 64 * 4096;   // LSTM pre-activations (reused)
  float* wHat  = ws + off; off += 64 * 1024;   // h_att
  float* wAh1  = ws + off; off += 64 * 512;
  float* wAh2  = ws + off; off += 64 * 512;
  float* wAh3  = ws + off; off += 64 * 512;
  float* wAtt1 = ws + off; off += 64 * 1024;
  float* wAtt2 = ws + off; off += 64 * 1024;
  float* wAtt3 = ws + off; off += 64 * 1024;
  float* wGt1  = ws + off; off += 64 * 1024;
  float* wGt2  = ws + off; off += 64 * 1024;
  float* wLn   = ws + off; off += 64 * 2048;
  float* wA23  = ws + off; off += 64 * 1024;
  float* wAtt  = ws + off; off += 64 * 1024;
  int*   wIdx  = (int*)(ws + off); off += 64;

  GArgs ga;

  // 1) att-LSTM pre-activations: [prev_h | fc | xt] @ Wih^T + h0 @ Whh^T + biases
  ga = {};
  ga.s[0] = { h1,       aWih + 0,    1024, 3072, 1024 };
  ga.s[1] = { fc_feats, aWih + 1024, 1024, 3072, 1024 };
  ga.s[2] = { xt,       aWih + 2048, 1024, 3072, 1024 };
  ga.s[3] = { h0,       aWhh,        1024, 1024, 1024 };
  ga.nseg = 4; ga.b1 = abih; ga.b2 = abhh; ga.C = wG; ga.N = 4096; ga.act = 0;
  gemm64_wmma_f32<<<32, 256, 0, stream>>>(ga);

  // 2) h_att, c_att
  lstm_pointwise<<<256, 256, 0, stream>>>(wG, c0, wHat, o_hatt_s, o_catt_s);

  // 3) three h2att projections (64x512, K=1024)
  ga = {}; ga.s[0] = { wHat, a1W, 1024, 1024, 1024 };
  ga.nseg = 1; ga.b1 = a1b; ga.b2 = nullptr; ga.C = wAh1; ga.N = 512; ga.act = 0;
  gemm64_wmma_f32<<<4, 256, 0, stream>>>(ga);
  ga.s[0].W = a2W; ga.b1 = a2b; ga.C = wAh2;
  gemm64_wmma_f32<<<4, 256, 0, stream>>>(ga);
  ga.s[0].W = a3W; ga.b1 = a3b; ga.C = wAh3;
  gemm64_wmma_f32<<<4, 256, 0, stream>>>(ga);

  // 4) attention 1 (+ argmax), then gathered attentions 2 and 3
  attention_kernel<<<64, 256, 0, stream>>>(wAh1, att_feats, p_att, att_masks,
                                           nullptr, a1aW, a1ab, wAtt1, wIdx);
  attention_kernel<<<64, 256, 0, stream>>>(wAh2, pos_feats, p_pos, pos_mask,
                                           wIdx, a2aW, a2ab, wAtt2, nullptr);
  attention_kernel<<<64, 256, 0, stream>>>(wAh3, sem_feats, p_sem, sem_mask,
                                           wIdx, a3aW, a3ab, wAtt3, nullptr);

  // 5) gates g1, g2 = sigmoid([xt | h_att] @ gW^T + gb)
  ga = {};
  ga.s[0] = { xt,   g1W,        1024, 2048, 1024 };
  ga.s[1] = { wHat, g1W + 1024, 1024, 2048, 1024 };
  ga.nseg = 2; ga.b1 = g1b; ga.b2 = nullptr; ga.C = wGt1; ga.N = 1024; ga.act = 1;
  gemm64_wmma_f32<<<8, 256, 0, stream>>>(ga);
  ga.s[0].W = g2W; ga.s[1].W = g2W + 1024; ga.b1 = g2b; ga.C = wGt2;
  gemm64_wmma_f32<<<8, 256, 0, stream>>>(ga);

  // 6) LN(g1*att2) || LN((1-g1)*att3) then @ att23_W^T
  ln_gate_kernel<<<64, 256, 0, stream>>>(wGt1, wAtt2, wAtt3, ln1g, ln1b, ln2g, ln2b, wLn);
  ga = {}; ga.s[0] = { wLn, att23W, 2048, 2048, 2048 };
  ga.nseg = 1; ga.b1 = nullptr; ga.b2 = nullptr; ga.C = wA23; ga.N = 1024; ga.act = 0;
  gemm64_wmma_f32<<<8, 256, 0, stream>>>(ga);

  // 7) att = g2*att1 + (1-g2)*att23
  combine_kernel<<<256, 256, 0, stream>>>(wGt2, wAtt1, wA23, wAtt);

  // 8) lang-LSTM pre-activations: [att | h_att] @ lWih^T + h_state[1] @ lWhh^T + biases
  ga = {};
  ga.s[0] = { wAtt, lWih + 0,    1024, 2048, 1024 };
  ga.s[1] = { wHat, lWih + 1024, 1024, 2048, 1024 };
  ga.s[2] = { h1,   lWhh,        1024, 1024, 1024 };
  ga.nseg = 3; ga.b1 = lbih; ga.b2 = lbhh; ga.C = wG; ga.N = 4096; ga.act = 0;
  gemm64_wmma_f32<<<32, 256, 0, stream>>>(ga);

  // 9) h_lang, c_lang -> output slots
  lstm_pointwise<<<256, 256, 0, stream>>>(wG, c1, o_hlang, o_hlang_s, o_clang_s);
}